// Nms3dSagittalForamina_63737314673342
// MI455X (gfx1250) — compile-verified
//
#include <hip/hip_runtime.h>

#define IOU_TR   0.3f
#define EPS_F    1e-7f
#define BLK      256        // threads per block (8 wave32 waves)
#define STAGES   4          // async-to-LDS ring depth in pass 2

// ---------- wave32 helpers ----------
__device__ __forceinline__ unsigned long long umax64(unsigned long long a, unsigned long long b) {
    return a > b ? a : b;
}

__device__ __forceinline__ unsigned long long shfl_xor_u64(unsigned long long v, int mask) {
    unsigned lo = (unsigned)(v & 0xFFFFFFFFull);
    unsigned hi = (unsigned)(v >> 32);
    lo = __shfl_xor(lo, mask, 32);   // wave32: width 32, masks 16..1
    hi = __shfl_xor(hi, mask, 32);
    return ((unsigned long long)hi << 32) | (unsigned long long)lo;
}

// pack (score, index): non-negative f32 bits are monotone as u32; ~idx makes
// equal scores prefer the LOWEST index, matching jnp.argmax's first-occurrence.
__device__ __forceinline__ unsigned long long pack_score_idx(float s, unsigned i) {
    return ((unsigned long long)__float_as_uint(s) << 32) |
           (unsigned long long)(unsigned)(~i);
}

// block-level reduction of two per-class packed maxima, then one atomic each
__device__ __forceinline__ void block_commit(unsigned long long b0, unsigned long long b1,
                                             unsigned long long* ws, int slot0, int slot1) {
    __shared__ unsigned long long s0[BLK / 32];
    __shared__ unsigned long long s1[BLK / 32];
    #pragma unroll
    for (int m = 16; m > 0; m >>= 1) {          // wave32 tree
        b0 = umax64(b0, shfl_xor_u64(b0, m));
        b1 = umax64(b1, shfl_xor_u64(b1, m));
    }
    const int lane = threadIdx.x & 31;
    const int wave = threadIdx.x >> 5;
    if (lane == 0) { s0[wave] = b0; s1[wave] = b1; }
    __syncthreads();
    if (threadIdx.x == 0) {
        unsigned long long r0 = s0[0], r1 = s1[0];
        for (int w = 1; w < BLK / 32; ++w) { r0 = umax64(r0, s0[w]); r1 = umax64(r1, s1[w]); }
        if (r0) atomicMax(ws + slot0, r0);
        if (r1) atomicMax(ws + slot1, r1);
    }
}

// ---------- kernels ----------
__global__ void nms_init(unsigned long long* ws) {
    if (threadIdx.x < 4) ws[threadIdx.x] = 0ull;
}

// Pass 1: per-class argmax of score. 8B (score,cls) loads; every 32B row's
// cache line lands in the 192MB L2 anyway, prefilling pass 2. Unroll x8:
// eight independent global_load_b64 in flight per thread.
__global__ void nms_pass1(const float* __restrict__ results, int n,
                          unsigned long long* __restrict__ ws) {
    const long long stride = (long long)gridDim.x * blockDim.x;
    unsigned long long b0 = 0, b1 = 0;

    auto acc = [&](float2 sc, long long i) {
        const float s = sc.x, c = sc.y;
        if (s >= 0.0f) {                       // SCORE_TR = 0.0
            const unsigned long long p = pack_score_idx(s, (unsigned)i);
            if (c == 0.0f)      b0 = umax64(b0, p);
            else if (c == 1.0f) b1 = umax64(b1, p);
        }
    };

    long long i = (long long)blockIdx.x * blockDim.x + threadIdx.x;
    for (; i + 7 * stride < (long long)n; i += 8 * stride) {
        __builtin_prefetch(results + (size_t)(i + 8 * stride) * 8, 0, 1); // global_prefetch_b8
        float2 r[8];
        #pragma unroll
        for (int u = 0; u < 8; ++u)           // 8 independent b64 loads, then consume
            r[u] = *reinterpret_cast<const float2*>(results + (size_t)(i + u * stride) * 8);
        #pragma unroll
        for (int u = 0; u < 8; ++u)
            acc(r[u], i + u * stride);
    }
    for (; i < (long long)n; i += stride) {
        acc(*reinterpret_cast<const float2*>(results + (size_t)i * 8), i);
    }
    block_commit(b0, b1, ws, 0, 1);
}

// Pass 2: IoU3D vs each class's b0, masked per-class argmax.
// Data is L2-resident from pass 1; streamed through a 4-stage
// GLOBAL_LOAD_ASYNC_TO_LDS_B128 ring (ASYNCcnt-tracked), consumed via
// ds_load_b128 — in-flight bytes live in LDS, not VGPRs.
__global__ void nms_pass2(const float* __restrict__ results, int n,
                          const unsigned long long* __restrict__ ws_in,
                          unsigned long long* __restrict__ ws_out) {
    unsigned ia = ~(unsigned)ws_in[0]; if (ia >= (unsigned)n) ia = 0;  // argmax(all -inf)==0
    unsigned ib = ~(unsigned)ws_in[1]; if (ib >= (unsigned)n) ib = 0;
    const float* ra = results + (size_t)ia * 8 + 2;
    const float* rb = results + (size_t)ib * 8 + 2;
    const float ax1 = ra[0], ay1 = ra[1], az1 = ra[2], ax2 = ra[3], ay2 = ra[4], az2 = ra[5];
    const float bx1 = rb[0], by1 = rb[1], bz1 = rb[2], bx2 = rb[3], by2 = rb[4], bz2 = rb[5];
    const float va = (ax2 - ax1) * (ay2 - ay1) * (az2 - az1);
    const float vb = (bx2 - bx1) * (by2 - by1) * (bz2 - bz1);

    unsigned long long b0 = 0, b1 = 0;

    auto acc = [&](float4 lo, float4 hi, long long i) {
        const float s = lo.x, c = lo.y;
        const float x1 = lo.z, y1 = lo.w, z1 = hi.x, x2 = hi.y, y2 = hi.z, z2 = hi.w;
        const bool isA = (c == 0.0f);
        const float rx1 = isA ? ax1 : bx1, ry1 = isA ? ay1 : by1, rz1 = isA ? az1 : bz1;
        const float rx2 = isA ? ax2 : bx2, ry2 = isA ? ay2 : by2, rz2 = isA ? az2 : bz2;
        const float rv  = isA ? va  : vb;
        const float dx = fmaxf(fminf(rx2, x2) - fmaxf(rx1, x1), 0.0f);
        const float dy = fmaxf(fminf(ry2, y2) - fmaxf(ry1, y1), 0.0f);
        const float dz = fmaxf(fminf(rz2, z2) - fmaxf(rz1, z1), 0.0f);
        const float inter = dx * dy * dz;
        const float v2 = (x2 - x1) * (y2 - y1) * (z2 - z1);
        const float iou = inter / (rv + v2 - inter + EPS_F);
        if (s >= 0.0f && iou <= IOU_TR) {
            const unsigned long long p = pack_score_idx(s, (unsigned)i);
            if (c == 0.0f)      b0 = umax64(b0, p);
            else if (c == 1.0f) b1 = umax64(b1, p);
        }
    };

    const long long total = (long long)gridDim.x * blockDim.x;
    const long long gid   = (long long)blockIdx.x * blockDim.x + threadIdx.x;
    const long long full  = (long long)n / total;          // uniform per-thread trip count
    const long long rem   = (long long)n - full * total;

    // LDS ring: STAGES x (BLK threads x 32B row) = 32KB of the 320KB WGP pool
    __shared__ __align__(32) float stage_buf[STAGES * BLK * 8];
    const unsigned stage_bytes = BLK * 32u;
    const unsigned my_lds0 = (unsigned)(uintptr_t)(&stage_buf[0]) + (unsigned)threadIdx.x * 32u;
    const unsigned long long base = (unsigned long long)(unsigned long long)results; (void)base;

    auto issue_stage = [&](long long st) {     // two async b128: one 32B row -> LDS
        const unsigned loff = my_lds0 + (unsigned)(st & (STAGES - 1)) * stage_bytes;
        const unsigned goff = (unsigned)(((unsigned long long)gid + (unsigned long long)st *
                                          (unsigned long long)total) * 32ull);
        asm volatile("global_load_async_to_lds_b128 %0, %1, %2\n\t"
                     "global_load_async_to_lds_b128 %0, %1, %2 offset:16"
                     :: "v"(loff), "v"(goff), "s"(results) : "memory");
    };

    // prologue: stages 0..min(3,full)-1
    for (long long s = 0; s < (full < 3 ? full : 3); ++s) issue_stage(s);

    for (long long k = 0; k < full; ++k) {
        if (k + 3 < full) {
            // retire prior ds_loads first: kills WAR on the ring slot we rewrite
            asm volatile("s_wait_dscnt 0x0" ::: "memory");
            issue_stage(k + 3);
            asm volatile("s_wait_asynccnt 0x6" ::: "memory");   // 3 stages still in flight
        } else if (k + 2 < full) {
            asm volatile("s_wait_asynccnt 0x4" ::: "memory");
        } else if (k + 1 < full) {
            asm volatile("s_wait_asynccnt 0x2" ::: "memory");
        } else {
            asm volatile("s_wait_asynccnt 0x0" ::: "memory");
        }
        const float4* sp = reinterpret_cast<const float4*>(
            &stage_buf[(size_t)((k & (STAGES - 1)) * BLK + threadIdx.x) * 8]);
        const float4 lo = sp[0];
        const float4 hi = sp[1];
        acc(lo, hi, gid + k * total);
    }

    if (gid < rem) {                            // tail rows: direct loads
        const long long i = full * total + gid;
        const float4 lo = *reinterpret_cast<const float4*>(results + (size_t)i * 8);
        const float4 hi = *reinterpret_cast<const float4*>(results + (size_t)i * 8 + 4);
        acc(lo, hi, i);
    }
    block_commit(b0, b1, ws_out, 0, 1);
}

// Gather: out rows = [c0:i0, c0:i1, c1:i0, c1:i1], 8 floats each.
__global__ void nms_gather(const float* __restrict__ results, int n,
                           const unsigned long long* __restrict__ ws,
                           float* __restrict__ out) {
    const int t = threadIdx.x;
    if (t >= 32) return;
    const int row = t >> 3, col = t & 7;
    const int slot = (row == 0) ? 0 : (row == 1) ? 2 : (row == 2) ? 1 : 3;
    unsigned idx = ~(unsigned)ws[slot];
    if (idx >= (unsigned)n) idx = 0;
    out[t] = results[(size_t)idx * 8 + col];
}

extern "C" void kernel_launch(void* const* d_in, const int* in_sizes, int n_in,
                              void* d_out, int out_size, void* d_ws, size_t ws_size,
                              hipStream_t stream) {
    const float* results = (const float*)d_in[0];
    const int n = in_sizes[0] / 8;                      // rows of (N, 8) table
    unsigned long long* ws = (unsigned long long*)d_ws; // ws[0..1]=pass1, ws[2..3]=pass2
    float* out = (float*)d_out;

    const int threads = BLK;   // 8 waves/block on wave32
    const int blocks  = 1024;  // 262144 threads -> ~11 rows/thread

    nms_init  <<<1,      64,      0, stream>>>(ws);
    nms_pass1 <<<blocks, threads, 0, stream>>>(results, n, ws);
    nms_pass2 <<<blocks, threads, 0, stream>>>(results, n, ws, ws + 2);
    nms_gather<<<1,      32,      0, stream>>>(results, n, ws, out);
}